// WeightedGCNLayer_48129403519265
// MI455X (gfx1250) — compile-verified
//
#include <hip/hip_runtime.h>
#include <hip/hip_bf16.h>

// ---------------------------------------------------------------------------
// WeightedGCNLayer on gfx1250 (MI455X)
//
//   deg = scatter_add_dst(w)                      (E atomics, L2-resident)
//   agg = scatter_add_dst( norm_e * H[src] )      (E*64 atomics, L2-resident)
//   out = agg @ W^T                               (fp32 WMMA 16x16x4)
//
// Reassociation: (H[src] @ W^T)*norm scattered == scatter(norm*H[src]) @ W^T.
// Cuts GEMM FLOPs E/N = 16x and removes the [E,64] message tensor entirely.
// H (25.6 MB) + agg (25.6 MB) + edges (19 MB) all fit in the 192 MB L2.
//
// Workspace layout (floats):
//   [0 .. N)                     deg
//   [DEG_PAD .. DEG_PAD+N*64)    agg      (DEG_PAD = N rounded up to 64)
// ---------------------------------------------------------------------------

typedef __attribute__((ext_vector_type(2))) float v2f;
typedef __attribute__((ext_vector_type(8))) float v8f;

#define DIM 64

// ---- zero workspace -------------------------------------------------------
__global__ void gcn_zero_kernel(float* __restrict__ p, int n) {
    int i = blockIdx.x * blockDim.x + threadIdx.x;
    if (i < n) p[i] = 0.0f;
}

// ---- weighted in-degree: deg[dst] += w  (one edge per lane) ---------------
__global__ void gcn_deg_kernel(const int* __restrict__ ei,
                               const float* __restrict__ ew,
                               float* __restrict__ deg, int E) {
    int e = blockIdx.x * blockDim.x + threadIdx.x;
    if (e < E) {
        atomicAdd(&deg[ei[E + e]], ew[e]);   // dst row of edge_index
    }
}

// ---- fused norm + gather(H[src]) + scatter-add into agg[dst] --------------
// One wave (32 lanes) per edge; each lane owns 2 of the 64 dims.
// The edge id is forced wave-uniform via readfirstlane so the per-edge
// scalars (src, dst, w, deg[src], deg[dst]) lower to s_load_b32 (SMEM path)
// instead of 32 redundant vector loads per wave.
__global__ void __launch_bounds__(256)
gcn_scatter_kernel(const float* __restrict__ H,
                   const int* __restrict__ ei,
                   const float* __restrict__ ew,
                   const float* __restrict__ deg,
                   float* __restrict__ agg, int E) {
    int e = __builtin_amdgcn_readfirstlane(blockIdx.x * 8 + (threadIdx.x >> 5));
    int lane = threadIdx.x & 31;
    if (e >= E) return;

    int src = ei[e];
    int dst = ei[E + e];
    float w  = ew[e];
    float ds = fmaxf(deg[src], 1e-6f);
    float dd = fmaxf(deg[dst], 1e-6f);
    float norm = w / fmaxf(sqrtf(ds * dd), 1e-6f);

    const float2* hrow = (const float2*)(H + (size_t)src * DIM);
    float2 h = hrow[lane];                           // dims 2*lane, 2*lane+1

    float* arow = agg + (size_t)dst * DIM + 2 * lane;
    atomicAdd(arow + 0, norm * h.x);
    atomicAdd(arow + 1, norm * h.y);
}

// ---- out = agg @ W^T via V_WMMA_F32_16X16X4_F32 ---------------------------
// Block = 128 threads = 4 waves. Each wave computes one 16x16 tile:
// rows [blockIdx.x*16, +16), cols [wave*16, +16). K loop: 16 steps of 4.
__global__ void __launch_bounds__(128)
gcn_gemm_wmma_kernel(const float* __restrict__ agg,
                     const float* __restrict__ Wm,   // [DIM][DIM] row-major
                     float* __restrict__ out, int Nn) {
    const int wave    = threadIdx.x >> 5;    // 0..3 -> column tile
    const int lane    = threadIdx.x & 31;
    const int half    = lane >> 4;           // 0: K pair {0,1}, 1: {2,3}
    const int l       = lane & 15;           // M (for A) / N (for B/C/D)
    const int rowBase = blockIdx.x * 16;
    const int colBase = wave * 16;

    // B fragments: B[k][n] = W[n][k]  (n = colBase + l along lanes)
    // lane covers K = kk*4 + 2*half + {0,1}; contiguous pair -> b64 loads.
    v2f bfrag[16];
#pragma unroll
    for (int kk = 0; kk < 16; ++kk) {
        const float* wp = Wm + (size_t)(colBase + l) * DIM + kk * 4 + 2 * half;
        bfrag[kk] = *(const v2f*)wp;
    }

    int rA = rowBase + l;
    if (rA >= Nn) rA = Nn - 1;               // keep EXEC all-ones for WMMA

    v8f c = {};
#pragma unroll
    for (int kk = 0; kk < 16; ++kk) {
        // A fragment: A[m][k] = agg[rowBase+m][kk*4 + k], m on lanes 0..15,
        // K pair selected by lane half (ISA 7.12.2 32-bit A 16x4 layout).
        const float* ap = agg + (size_t)rA * DIM + kk * 4 + 2 * half;
        v2f a = *(const v2f*)ap;
        c = __builtin_amdgcn_wmma_f32_16x16x4_f32(
                /*neg_a=*/false, a, /*neg_b=*/false, bfrag[kk],
                /*c_mod=*/(short)0, c, /*reuse_a=*/false, /*reuse_b=*/false);
    }

    // D layout: VGPR r = row (r + 8*half), col = colBase + l
#pragma unroll
    for (int r = 0; r < 8; ++r) {
        int row = rowBase + r + 8 * half;
        if (row < Nn) out[(size_t)row * DIM + colBase + l] = c[r];
    }
}

// ---------------------------------------------------------------------------
extern "C" void kernel_launch(void* const* d_in, const int* in_sizes, int n_in,
                              void* d_out, int out_size, void* d_ws, size_t ws_size,
                              hipStream_t stream) {
    const float* H   = (const float*)d_in[0];        // [N, 64]
    const int*   ei  = (const int*)d_in[1];          // [2, E]
    const float* ew  = (const float*)d_in[2];        // [E]
    const float* Wm  = (const float*)d_in[3];        // [64, 64]
    float*       out = (float*)d_out;                // [N, 64]

    const int N = in_sizes[0] / DIM;
    const int E = in_sizes[2];

    const int degPad = ((N + 63) / 64) * 64;         // 256B-align agg
    float* deg = (float*)d_ws;
    float* agg = deg + degPad;

    // 1) zero deg + agg
    {
        int n = degPad + N * DIM;
        gcn_zero_kernel<<<(n + 255) / 256, 256, 0, stream>>>(deg, n);
    }
    // 2) weighted in-degree
    gcn_deg_kernel<<<(E + 255) / 256, 256, 0, stream>>>(ei, ew, deg, E);

    // 3) fused norm + gather + scatter-add (one wave per edge, 8 edges/block)
    gcn_scatter_kernel<<<(E + 7) / 8, 256, 0, stream>>>(H, ei, ew, deg, agg, E);

    // 4) out = agg @ W^T with fp32 WMMA
    gcn_gemm_wmma_kernel<<<(N + 15) / 16, 128, 0, stream>>>(agg, Wm, out, N);
}